// FusionBlock_25950192402744
// MI455X (gfx1250) — compile-verified
//
#include <hip/hip_runtime.h>
#include <math.h>

typedef __attribute__((ext_vector_type(2))) float v2f;
typedef __attribute__((ext_vector_type(8))) float v8f;

#define M_CTX   256
#define N_NODES 2048
#define D2      256
#define D4      512
#define LQ      64

// workspace layout (float offsets); total ~6.33 MB
#define WS_ENT  0
#define WS_HID  (N_NODES * D4)
#define WS_QV   (WS_HID + N_NODES * D2)
#define WS_HV1  (WS_QV + D4)
#define WS_HV2  (WS_HV1 + N_NODES)
#define WS_MX   (WS_HV2 + N_NODES)
#define WS_SM   (WS_MX + N_NODES)

// ---------------------------------------------------------------------------
// 1. ent = [mean_p | max_p] : fused masked mean + masked max over M context rows
//    block n (2048 blocks x 256 threads); bin column cached in LDS.
__global__ void ent_kernel(const float* __restrict__ c, const int* __restrict__ binM,
                           float* __restrict__ ent) {
  __shared__ float sbin[M_CTX];
  const int n = blockIdx.x;
  const int t = threadIdx.x;
  sbin[t] = (float)binM[t * N_NODES + n];
  __syncthreads();
  float sum = 0.f, mx = -INFINITY;
  for (int m = 0; m < M_CTX; ++m) {
    float v = c[m * D2 + t] * sbin[m];
    sum += v;
    mx = fmaxf(mx, v);
  }
  ent[n * D4 + t]      = sum * (1.0f / M_CTX);
  ent[n * D4 + D2 + t] = mx;
}

// ---------------------------------------------------------------------------
// 2. qV = mean(query_emb, axis=0) @ V   (1 block x 512 threads)
__global__ void qv_kernel(const float* __restrict__ query, const float* __restrict__ V,
                          float* __restrict__ qV) {
  __shared__ float qs[D2];
  const int t = threadIdx.x;  // 0..511
  if (t < D2) {
    float s = 0.f;
    for (int l = 0; l < LQ; ++l) s += query[l * D2 + t];
    qs[t] = s * (1.0f / LQ);
  }
  __syncthreads();
  float acc = 0.f;
  for (int d = 0; d < D2; ++d) acc += qs[d] * V[d * D4 + t];
  qV[t] = acc;
}

// ---------------------------------------------------------------------------
// 3. gamma_n = dot(qV, ent[n]) / sqrt(D2);  E[n] = sigmoid(gamma_n) * ent[n]  (in place)
__global__ void escale_kernel(float* __restrict__ ent, const float* __restrict__ qV) {
  __shared__ float red[D2];
  const int n = blockIdx.x, t = threadIdx.x;
  float e0 = ent[n * D4 + t], e1 = ent[n * D4 + D2 + t];
  red[t] = qV[t] * e0 + qV[D2 + t] * e1;
  __syncthreads();
  for (int s = 128; s > 0; s >>= 1) {
    if (t < s) red[t] += red[t + s];
    __syncthreads();
  }
  float gamma = red[0] * (1.0f / 16.0f);  // 1/sqrt(256)
  float sig = 1.0f / (1.0f + expf(-gamma));
  ent[n * D4 + t]      = e0 * sig;
  ent[n * D4 + D2 + t] = e1 * sig;
}

// ---------------------------------------------------------------------------
// 4. hidden = E @ U^T + b  via V_WMMA_F32_16X16X4_F32. One wave per 16x16 tile.
//    A pair (K=+0,+1 / +2,+3) and B pair are contiguous floats -> b64 loads.
__global__ void hidden_kernel(const float* __restrict__ E, const float* __restrict__ U,
                              const float* __restrict__ bvec, float* __restrict__ hidden) {
  const int lane = threadIdx.x;        // 32 threads = one wave
  const int i0 = blockIdx.x * 16;
  const int d0 = blockIdx.y * 16;
  const int lm = lane & 15;
  const int koff = (lane >> 4) << 1;   // lanes 0-15: K=0,1 ; lanes 16-31: K=2,3
  const float* arow = E + (i0 + lm) * D4;
  const float* brow = U + (d0 + lm) * D4;  // B[k,d] = U[d,k]
  v8f acc = {};
  for (int k = 0; k < D4; k += 4) {
    v2f a = *reinterpret_cast<const v2f*>(arow + k + koff);
    v2f b = *reinterpret_cast<const v2f*>(brow + k + koff);
    acc = __builtin_amdgcn_wmma_f32_16x16x4_f32(false, a, false, b, (short)0, acc,
                                                false, false);
  }
  const float bd = bvec[d0 + lm];
  const int ibase = i0 + ((lane >> 4) << 3);  // D rows: M=r (lanes<16), M=8+r (lanes>=16)
#pragma unroll
  for (int r = 0; r < 8; ++r)
    hidden[(ibase + r) * D2 + d0 + lm] = acc[r] + bd;
}

// ---------------------------------------------------------------------------
// 5. hv1 = hidden @ w1, hv2 = hidden @ w2  (block per row, dual LDS reduction)
__global__ void hv_kernel(const float* __restrict__ hidden, const float* __restrict__ W,
                          float* __restrict__ hv1, float* __restrict__ hv2) {
  __shared__ float r1[D2], r2[D2];
  const int i = blockIdx.x, t = threadIdx.x;
  float h = hidden[i * D2 + t];
  r1[t] = h * W[t];
  r2[t] = h * W[D2 + t];
  __syncthreads();
  for (int s = 128; s > 0; s >>= 1) {
    if (t < s) { r1[t] += r1[t + s]; r2[t] += r2[t + s]; }
    __syncthreads();
  }
  if (t == 0) { hv1[i] = r1[0]; hv2[i] = r2[0]; }
}

// ---------------------------------------------------------------------------
// 6. Row softmax stats over masked betas (masked entries ARE zeros, not -inf).
//    betas[i,j] = adj[i,j] ? lrelu(hv1[i]+hv2[j]) : 0 ; mx[i]=rowmax, sm[i]=sum exp(b-mx)
__global__ void softmax_stats_kernel(const int* __restrict__ adj,
                                     const float* __restrict__ hv1,
                                     const float* __restrict__ hv2,
                                     float* __restrict__ mx, float* __restrict__ sm) {
  __shared__ float red[D2];
  const int i = blockIdx.x, t = threadIdx.x;
  const float h1 = hv1[i];
  float beta[8];
  float lmx = -INFINITY;
#pragma unroll
  for (int s = 0; s < 8; ++s) {
    int j = t + s * D2;
    float x = h1 + hv2[j];
    float b = (x >= 0.f) ? x : 0.01f * x;   // jax.nn.leaky_relu default slope
    b = adj[i * N_NODES + j] ? b : 0.0f;
    beta[s] = b;
    lmx = fmaxf(lmx, b);
  }
  red[t] = lmx;
  __syncthreads();
  for (int s = 128; s > 0; s >>= 1) {
    if (t < s) red[t] = fmaxf(red[t], red[t + s]);
    __syncthreads();
  }
  float rowmax = red[0];
  __syncthreads();
  float lsum = 0.f;
#pragma unroll
  for (int s = 0; s < 8; ++s) lsum += expf(beta[s] - rowmax);
  red[t] = lsum;
  __syncthreads();
  for (int s = 128; s > 0; s >>= 1) {
    if (t < s) red[t] += red[t + s];
    __syncthreads();
  }
  if (t == 0) { mx[i] = rowmax; sm[i] = red[0]; }
}

// ---------------------------------------------------------------------------
// 7. out = relu(msg_w @ hidden)^T. msg_w tiles built on the fly:
//    msg_w[i,j] = adj[i,j] ? exp((adj[j,i]?lrelu(hv1[j]+hv2[i]):0) - mx[j])/sm[j] : 0
//    4 waves/block; each wave owns one i-tile x ALL 16 d-tiles (128 acc VGPRs)
//    so every msg_w element (2 adj loads + exp) is computed exactly once.
//    hidden staged through LDS in 32x256-float chunks (32 KB).
#define KCH 32
__global__ void __launch_bounds__(128) out_kernel(
    const float* __restrict__ hidden, const int* __restrict__ adj,
    const float* __restrict__ hv1, const float* __restrict__ hv2,
    const float* __restrict__ mx, const float* __restrict__ sm,
    float* __restrict__ out) {
  __shared__ float hs[KCH * D2];
  const int tid = threadIdx.x;
  const int wave = tid >> 5;
  const int lane = tid & 31;
  const int lm = lane & 15;
  const int half = lane >> 4;              // 0 or 1 -> K pair +0/+1 vs +2/+3
  const int i0 = blockIdx.x * 64 + wave * 16;
  const int irow = i0 + lm;
  const float h2i = hv2[irow];

  v8f zero = {};
  v8f acc[16];
#pragma unroll
  for (int t = 0; t < 16; ++t) acc[t] = zero;

  for (int kc = 0; kc < N_NODES; kc += KCH) {
    __syncthreads();
    // cooperative coalesced copy: hidden[kc:kc+32, :] -> LDS (float4)
    const float4* src4 = reinterpret_cast<const float4*>(hidden + kc * D2);
    float4* dst4 = reinterpret_cast<float4*>(hs);
#pragma unroll
    for (int r = 0; r < 16; ++r) dst4[r * 128 + tid] = src4[r * 128 + tid];
    __syncthreads();

    for (int kk = 0; kk < KCH; kk += 4) {
      const int jb = kc + kk + half * 2;
      float av[2];
#pragma unroll
      for (int u = 0; u < 2; ++u) {
        const int j = jb + u;
        float val = 0.f;
        if (adj[irow * N_NODES + j]) {
          float x = hv1[j] + h2i;
          float b = (x >= 0.f) ? x : 0.01f * x;
          b = adj[j * N_NODES + irow] ? b : 0.0f;
          val = expf(b - mx[j]) / sm[j];
        }
        av[u] = val;
      }
      v2f a; a[0] = av[0]; a[1] = av[1];
      const int r0 = (kk + half * 2) * D2 + lm;
#pragma unroll
      for (int t = 0; t < 16; ++t) {
        v2f b;
        b[0] = hs[r0 + t * 16];
        b[1] = hs[r0 + D2 + t * 16];
        acc[t] = __builtin_amdgcn_wmma_f32_16x16x4_f32(false, a, false, b, (short)0,
                                                       acc[t], false, false);
      }
    }
  }

  // relu + transposed store: out[d, i] = relu(E_t[i, d])
  const int ibase = i0 + (half << 3);
#pragma unroll
  for (int t = 0; t < 16; ++t) {
    const int d = t * 16 + lm;
#pragma unroll
    for (int r = 0; r < 8; ++r)
      out[d * N_NODES + ibase + r] = fmaxf(acc[t][r], 0.0f);
  }
}

// ---------------------------------------------------------------------------
extern "C" void kernel_launch(void* const* d_in, const int* in_sizes, int n_in,
                              void* d_out, int out_size, void* d_ws, size_t ws_size,
                              hipStream_t stream) {
  const float* ctx   = (const float*)d_in[0];   // (1,256,256)
  const float* query = (const float*)d_in[1];   // (64,256)
  const float* V     = (const float*)d_in[2];   // (256,512)
  const float* U     = (const float*)d_in[3];   // (256,512)
  const float* bvec  = (const float*)d_in[4];   // (256,)
  const float* W     = (const float*)d_in[5];   // (512,1)
  const int*   binM  = (const int*)d_in[6];     // (256,2048)
  const int*   adj   = (const int*)d_in[7];     // (2048,2048)
  float* ws  = (float*)d_ws;
  float* ent = ws + WS_ENT;   // 2048x512, becomes E in place
  float* hid = ws + WS_HID;   // 2048x256
  float* qV  = ws + WS_QV;
  float* hv1 = ws + WS_HV1;
  float* hv2 = ws + WS_HV2;
  float* mxv = ws + WS_MX;
  float* smv = ws + WS_SM;
  float* out = (float*)d_out; // (256,2048)

  ent_kernel<<<N_NODES, D2, 0, stream>>>(ctx, binM, ent);
  qv_kernel<<<1, D4, 0, stream>>>(query, V, qV);
  escale_kernel<<<N_NODES, D2, 0, stream>>>(ent, qV);
  hidden_kernel<<<dim3(N_NODES / 16, D2 / 16), 32, 0, stream>>>(ent, U, bvec, hid);
  hv_kernel<<<N_NODES, D2, 0, stream>>>(hid, W, hv1, hv2);
  softmax_stats_kernel<<<N_NODES, D2, 0, stream>>>(adj, hv1, hv2, mxv, smv);
  out_kernel<<<N_NODES / 64, 128, 0, stream>>>(hid, adj, hv1, hv2, mxv, smv, out);
}